// FeatGuide_Drop_66606352827273
// MI455X (gfx1250) — compile-verified
//
#include <hip/hip_runtime.h>
#include <math.h>

typedef float v2f __attribute__((ext_vector_type(2)));
typedef float v8f __attribute__((ext_vector_type(8)));

#define BSZ  16
#define CCH  256
#define RR   16
#define HW   16384   // 128*128

// ---------------------------------------------------------------------------
// K1: s[b][c] = mean over HW of x[b][c][:]   (one block per (b,c) plane)
// ---------------------------------------------------------------------------
__global__ void k1_chan_mean(const float* __restrict__ x, float* __restrict__ s) {
    const size_t plane = (size_t)blockIdx.x * HW;
    const float4* xp = (const float4*)(x + plane);
    float sum = 0.f;
    for (int i = threadIdx.x; i < HW / 4; i += 256) {
        float4 v = xp[i];
        sum += v.x + v.y + v.z + v.w;
    }
    __shared__ float red[256];
    red[threadIdx.x] = sum;
    __syncthreads();
    for (int off = 128; off > 0; off >>= 1) {
        if (threadIdx.x < off) red[threadIdx.x] += red[threadIdx.x + off];
        __syncthreads();
    }
    if (threadIdx.x == 0) s[blockIdx.x] = red[0] * (1.0f / (float)HW);
}

// ---------------------------------------------------------------------------
// K2: SE MLP entirely in one wave with V_WMMA_F32_16X16X4_F32.
//   GEMM1: T = relu(s[16x256] @ w1[256x16])       (64 k-steps of 4)
//   GEMM2: gate = sigmoid(T[16x16] @ w2[16x256])  (16 N-tiles, 4 k-steps)
// Fragment layouts per CDNA5 ISA 7.12.2:
//   A (16x4 f32): lane 0-15 -> M=lane, VGPR0=K(k0+0), VGPR1=K(k0+1);
//                 lane16-31 -> M=lane-16, VGPR0=K(k0+2), VGPR1=K(k0+3)
//   B (4x16 f32): lane holds N=lane&15; VGPR0 row K(k0 + 0/2), VGPR1 K(k0 + 1/3)
//   C/D (16x16 f32): VGPR j = row j (lanes 0-15) / row j+8 (lanes 16-31), N=lane&15
// ---------------------------------------------------------------------------
__global__ void k2_se_mlp(const float* __restrict__ s,
                          const float* __restrict__ w1,
                          const float* __restrict__ w2,
                          float* __restrict__ gate) {
    const int lane = threadIdx.x;          // 0..31, all lanes active
    const int mn   = lane & 15;
    const int kh   = (lane >> 4) << 1;     // 0 for lanes 0-15, 2 for 16-31

    // ---- GEMM1: [16x256] x [256x16] -> [16x16] ----
    v8f acc = {};
    for (int k0 = 0; k0 < CCH; k0 += 4) {
        v2f a, b;
        a.x = s[mn * CCH + k0 + kh];
        a.y = s[mn * CCH + k0 + kh + 1];
        b.x = w1[(k0 + kh)     * RR + mn];
        b.y = w1[(k0 + kh + 1) * RR + mn];
        acc = __builtin_amdgcn_wmma_f32_16x16x4_f32(
            /*neg_a=*/false, a, /*neg_b=*/false, b,
            /*c_mod=*/(short)0, acc, /*reuse_a=*/false, /*reuse_b=*/false);
    }

    // relu, stage through LDS to convert D-layout -> A-layout
    __shared__ float T[16][16];
#pragma unroll
    for (int j = 0; j < 8; ++j) {
        const int mrow = (lane < 16) ? j : (j + 8);
        T[mrow][mn] = fmaxf(acc[j], 0.0f);
    }
    __syncthreads();

    // ---- GEMM2: [16x16] x [16x256] -> [16x256], 16 tiles of N=16 ----
    for (int nt = 0; nt < 16; ++nt) {
        v8f acc2 = {};
#pragma unroll
        for (int k0 = 0; k0 < 16; k0 += 4) {
            v2f a, b;
            a.x = T[mn][k0 + kh];
            a.y = T[mn][k0 + kh + 1];
            b.x = w2[(k0 + kh)     * CCH + nt * 16 + mn];
            b.y = w2[(k0 + kh + 1) * CCH + nt * 16 + mn];
            acc2 = __builtin_amdgcn_wmma_f32_16x16x4_f32(
                false, a, false, b, (short)0, acc2, false, false);
        }
#pragma unroll
        for (int j = 0; j < 8; ++j) {
            const int mrow = (lane < 16) ? j : (j + 8);
            const float g  = 1.0f / (1.0f + __expf(-acc2[j]));
            gate[mrow * CCH + nt * 16 + mn] = g;   // gate[b][c]
        }
    }
}

// ---------------------------------------------------------------------------
// K3: m[b][p] = (1/C) * sum_c x[b][c][p] * gate[b][c]
// 256 blocks: 16 batches x 16 pixel-chunks of 1024; float4 per thread.
// ---------------------------------------------------------------------------
__global__ void k3_gated_mean(const float* __restrict__ x,
                              const float* __restrict__ gate,
                              float* __restrict__ m) {
    const int b     = blockIdx.x >> 4;
    const int chunk = blockIdx.x & 15;
    const int p     = chunk * 1024 + threadIdx.x * 4;

    __shared__ float g[CCH];
    g[threadIdx.x] = gate[b * CCH + threadIdx.x];
    __syncthreads();

    const float* xb = x + (size_t)b * CCH * HW;
    float4 acc = {0.f, 0.f, 0.f, 0.f};
    for (int c = 0; c < CCH; ++c) {
        const float4 v = *(const float4*)(xb + (size_t)c * HW + p);
        const float gc = g[c];
        acc.x += v.x * gc; acc.y += v.y * gc;
        acc.z += v.z * gc; acc.w += v.w * gc;
    }
    const float inv = 1.0f / (float)CCH;
    float4 r = {acc.x * inv, acc.y * inv, acc.z * inv, acc.w * inv};
    *(float4*)(m + (size_t)b * HW + p) = r;
}

// ---------------------------------------------------------------------------
// K4: per batch: f = sigmoid(boxsum3x3(m)/9); thr = w_i*var(f,ddof=1)+mean(f);
//     mask[b] = (1 > thr) ? 0 : 1.   One block per batch.
// ---------------------------------------------------------------------------
__global__ void k4_mask(const float* __restrict__ m,
                        const float* __restrict__ w_i,
                        float* __restrict__ mask) {
    const int b = blockIdx.x;
    const float* mb = m + (size_t)b * HW;
    float sum = 0.f, sumsq = 0.f;
    for (int p = threadIdx.x; p < HW; p += 256) {
        const int y = p >> 7, xx = p & 127;
        float box = 0.f;
#pragma unroll
        for (int dy = -1; dy <= 1; ++dy) {
            const int yy = y + dy;
            if (yy < 0 || yy > 127) continue;
#pragma unroll
            for (int dx = -1; dx <= 1; ++dx) {
                const int xc = xx + dx;
                if (xc < 0 || xc > 127) continue;
                box += mb[yy * 128 + xc];
            }
        }
        const float f = 1.0f / (1.0f + __expf(-box * (1.0f / 9.0f)));
        sum += f; sumsq += f * f;
    }
    __shared__ float rs[256], rq[256];
    rs[threadIdx.x] = sum; rq[threadIdx.x] = sumsq;
    __syncthreads();
    for (int off = 128; off > 0; off >>= 1) {
        if (threadIdx.x < off) {
            rs[threadIdx.x] += rs[threadIdx.x + off];
            rq[threadIdx.x] += rq[threadIdx.x + off];
        }
        __syncthreads();
    }
    if (threadIdx.x == 0) {
        const float n    = (float)HW;
        const float mean = rs[0] / n;
        const float var  = (rq[0] - n * mean * mean) / (n - 1.0f);
        const float thr  = w_i[0] * var + mean;
        mask[b] = (1.0f > thr) ? 0.0f : 1.0f;
    }
}

// ---------------------------------------------------------------------------
// K5: out = x * gate[b][c] * mask[b].  One block = 1024 elems in one plane.
// ---------------------------------------------------------------------------
__global__ void k5_apply(const float* __restrict__ x,
                         const float* __restrict__ gate,
                         const float* __restrict__ mask,
                         float* __restrict__ out) {
    const int plane = blockIdx.x >> 4;     // b*256 + c
    const int chunk = blockIdx.x & 15;
    const int b     = plane >> 8;
    const float gm  = gate[plane] * mask[b];
    const size_t base = (size_t)plane * HW + (size_t)chunk * 1024 + threadIdx.x * 4;
    const float4 v = *(const float4*)(x + base);
    float4 r = {v.x * gm, v.y * gm, v.z * gm, v.w * gm};
    *(float4*)(out + base) = r;
}

// ---------------------------------------------------------------------------
extern "C" void kernel_launch(void* const* d_in, const int* in_sizes, int n_in,
                              void* d_out, int out_size, void* d_ws, size_t ws_size,
                              hipStream_t stream) {
    const float* x  = (const float*)d_in[0];   // [16,256,128,128]
    const float* w1 = (const float*)d_in[1];   // [256,16]
    const float* w2 = (const float*)d_in[2];   // [16,256]
    const float* wi = (const float*)d_in[3];   // [1]
    float* out = (float*)d_out;

    float* ws   = (float*)d_ws;
    float* s    = ws;                    // 4096 floats
    float* gate = ws + 4096;             // 4096 floats
    float* m    = ws + 8192;             // 262144 floats
    float* mask = ws + 8192 + 262144;    // 16 floats   (~1.03 MB total)

    k1_chan_mean<<<BSZ * CCH, 256, 0, stream>>>(x, s);
    k2_se_mlp  <<<1, 32, 0, stream>>>(s, w1, w2, gate);
    k3_gated_mean<<<BSZ * 16, 256, 0, stream>>>(x, gate, m);
    k4_mask    <<<BSZ, 256, 0, stream>>>(m, wi, mask);
    k5_apply   <<<BSZ * CCH * 16, 256, 0, stream>>>(x, gate, mask, out);
}